// GlobalAttentionLayer_88862873354915
// MI455X (gfx1250) — compile-verified
//
#include <hip/hip_runtime.h>

typedef __attribute__((ext_vector_type(2))) float v2f;
typedef __attribute__((ext_vector_type(8))) float v8f;

#define IN_F 128
#define OUT_F 48
#define LDS_STRIDE 132   // 128 + 4 pad: conflict-free b64 LDS reads across the wave

// ---------------------------------------------------------------------------
// Kernel 1: h = features @ W^T + b   via V_WMMA_F32_16X16X4_F32 (full fp32)
// One wave -> 16 rows x 48 cols. W (48x128 = 24KB) staged in LDS once/block.
// ---------------------------------------------------------------------------
__global__ void __launch_bounds__(256)
gat_gemm_wmma(const float* __restrict__ feat, const float* __restrict__ W,
              const float* __restrict__ bias, float* __restrict__ h, int n_nodes)
{
    __shared__ float Wlds[OUT_F * LDS_STRIDE];
    for (int i = threadIdx.x; i < OUT_F * IN_F; i += blockDim.x) {
        int r = i >> 7, c = i & (IN_F - 1);
        Wlds[r * LDS_STRIDE + c] = W[i];
    }
    __syncthreads();

    const int wave = threadIdx.x >> 5;
    const int lane = threadIdx.x & 31;
    const int half = lane >> 4;     // K-half select (A/B layout)
    const int lrow = lane & 15;     // M for A, N for B

    const int rowBase = (blockIdx.x * 8 + wave) * 16;
    if (rowBase >= n_nodes) return;           // uniform per wave

    int arow = rowBase + lrow;
    if (arow >= n_nodes) arow = n_nodes - 1;  // clamp tail loads (masked on store)

    const float* aptr = feat + (long long)arow * IN_F + half * 2;
    const float* b0p  = &Wlds[(lrow +  0) * LDS_STRIDE + half * 2];
    const float* b1p  = &Wlds[(lrow + 16) * LDS_STRIDE + half * 2];
    const float* b2p  = &Wlds[(lrow + 32) * LDS_STRIDE + half * 2];

    v8f acc0 = {}, acc1 = {}, acc2 = {};
#pragma unroll 8
    for (int k = 0; k < IN_F; k += 4) {
        v2f a  = *(const v2f*)(aptr + k);
        v2f b0 = *(const v2f*)(b0p + k);
        v2f b1 = *(const v2f*)(b1p + k);
        v2f b2 = *(const v2f*)(b2p + k);
        acc0 = __builtin_amdgcn_wmma_f32_16x16x4_f32(false, a, false, b0, (short)0, acc0, false, false);
        acc1 = __builtin_amdgcn_wmma_f32_16x16x4_f32(false, a, false, b1, (short)0, acc1, false, false);
        acc2 = __builtin_amdgcn_wmma_f32_16x16x4_f32(false, a, false, b2, (short)0, acc2, false, false);
    }

    const float bb0 = bias[lrow], bb1 = bias[lrow + 16], bb2 = bias[lrow + 32];
#pragma unroll
    for (int r = 0; r < 8; ++r) {
        int m = rowBase + r + half * 8;       // C/D layout: vgpr r, halves hold M=r / M=r+8
        if (m < n_nodes) {
            float* hp = h + (long long)m * OUT_F + lrow;
            hp[0]  = acc0[r] + bb0;
            hp[16] = acc1[r] + bb1;
            hp[32] = acc2[r] + bb2;
        }
    }
}

// ---------------------------------------------------------------------------
// Kernel 2: per-node attention scores s_src = h.w[:48], s_dst = h.w[48:]
// ---------------------------------------------------------------------------
__global__ void gat_scores(const float* __restrict__ h, const float* __restrict__ attn_w,
                           float* __restrict__ s_src, float* __restrict__ s_dst, int n)
{
    int i = blockIdx.x * blockDim.x + threadIdx.x;
    if (i >= n) return;
    const float* hp = h + (long long)i * OUT_F;
    float a = 0.f, b = 0.f;
#pragma unroll
    for (int f = 0; f < OUT_F; ++f) {
        float v = hp[f];
        a += v * attn_w[f];
        b += v * attn_w[OUT_F + f];
    }
    s_src[i] = a;
    s_dst[i] = b;
}

// ---------------------------------------------------------------------------
// Kernel 0: init out=0, segment-max keys=INT_MIN, denom=0
// ---------------------------------------------------------------------------
__global__ void gat_init(float* __restrict__ out, int n_nodes,
                         int* __restrict__ m_bits, float* __restrict__ denom)
{
    long long idx = (long long)blockIdx.x * blockDim.x + threadIdx.x;
    long long total = (long long)n_nodes * OUT_F;
    if (idx < total) out[idx] = 0.f;
    if (idx < n_nodes) { m_bits[idx] = (int)0x80000000; denom[idx] = 0.f; }
}

__device__ __forceinline__ int f32_to_ordered(float f) {
    int b = __float_as_int(f);
    return (b >= 0) ? b : (b ^ 0x7fffffff);
}
__device__ __forceinline__ float ordered_to_f32(int k) {
    return (k >= 0) ? __int_as_float(k) : __int_as_float(k ^ 0x7fffffff);
}

// ---------------------------------------------------------------------------
// Kernel 3: edge score + leaky-relu + segment max (ordered-int atomicMax)
// ---------------------------------------------------------------------------
__global__ void gat_edge_max(const float* __restrict__ s_src, const float* __restrict__ s_dst,
                             const int* __restrict__ src, const int* __restrict__ dst,
                             const float* __restrict__ attn_b,
                             float* __restrict__ e_buf, int* __restrict__ m_bits, int n_edges)
{
    int i = blockIdx.x * blockDim.x + threadIdx.x;
    if (i >= n_edges) return;
    float e = s_src[src[i]] + s_dst[dst[i]] + attn_b[0];
    e = (e < 0.f) ? 0.2f * e : e;              // leaky_relu(0.2)
    e_buf[i] = e;
    atomicMax(&m_bits[dst[i]], f32_to_ordered(e));
}

// ---------------------------------------------------------------------------
// Kernel 4: ex = exp(e - m[dst]); denom[dst] += ex  (e_buf overwritten w/ ex)
// ---------------------------------------------------------------------------
__global__ void gat_edge_exp(float* __restrict__ e_buf, const int* __restrict__ dst,
                             const int* __restrict__ m_bits, float* __restrict__ denom, int n_edges)
{
    int i = blockIdx.x * blockDim.x + threadIdx.x;
    if (i >= n_edges) return;
    int d = dst[i];
    float m = ordered_to_f32(m_bits[d]);
    float ex = __expf(e_buf[i] - m);
    e_buf[i] = ex;
    atomicAdd(&denom[d], ex);
}

// ---------------------------------------------------------------------------
// Kernel 5: alpha = ex / denom[dst]  (in place)
// ---------------------------------------------------------------------------
__global__ void gat_alpha(float* __restrict__ e_buf, const float* __restrict__ denom,
                          const int* __restrict__ dst, int n_edges)
{
    int i = blockIdx.x * blockDim.x + threadIdx.x;
    if (i >= n_edges) return;
    e_buf[i] = e_buf[i] / denom[dst[i]];
}

// ---------------------------------------------------------------------------
// Kernel 6: out[dst] += h[src] * alpha  — one thread per (edge, feature);
// 48 consecutive threads share an edge => coalesced h reads / out atomics.
// ---------------------------------------------------------------------------
__global__ void gat_scatter(const float* __restrict__ h, const int* __restrict__ src,
                            const int* __restrict__ dst, const float* __restrict__ alpha,
                            float* __restrict__ out, long long total)
{
    long long idx = (long long)blockIdx.x * blockDim.x + threadIdx.x;
    if (idx >= total) return;
    int e = (int)(idx / OUT_F);
    int f = (int)(idx - (long long)e * OUT_F);
    float v = h[(long long)src[e] * OUT_F + f] * alpha[e];
    atomicAdd(&out[(long long)dst[e] * OUT_F + f], v);
}

// ---------------------------------------------------------------------------
extern "C" void kernel_launch(void* const* d_in, const int* in_sizes, int n_in,
                              void* d_out, int out_size, void* d_ws, size_t ws_size,
                              hipStream_t stream)
{
    const float* feat   = (const float*)d_in[0];
    const float* W_w    = (const float*)d_in[1];
    const float* W_b    = (const float*)d_in[2];
    const float* attn_w = (const float*)d_in[3];
    const float* attn_b = (const float*)d_in[4];
    const int*   src    = (const int*)d_in[5];
    const int*   dst    = (const int*)d_in[6];

    const int n_nodes = in_sizes[0] / IN_F;
    const int n_edges = in_sizes[5];

    char* ws = (char*)d_ws;
    size_t off = 0;
    auto alloc = [&](size_t bytes) -> void* {
        void* p = ws + off;
        off += bytes;
        off = (off + 255) & ~(size_t)255;
        return p;
    };
    float* h      = (float*)alloc((size_t)n_nodes * OUT_F * sizeof(float));
    float* s_src  = (float*)alloc((size_t)n_nodes * sizeof(float));
    float* s_dst  = (float*)alloc((size_t)n_nodes * sizeof(float));
    float* e_buf  = (float*)alloc((size_t)n_edges * sizeof(float));
    int*   m_bits = (int*)  alloc((size_t)n_nodes * sizeof(int));
    float* denom  = (float*)alloc((size_t)n_nodes * sizeof(float));

    float* out = (float*)d_out;

    long long out_count = (long long)n_nodes * OUT_F;
    gat_init<<<(int)((out_count + 255) / 256), 256, 0, stream>>>(out, n_nodes, m_bits, denom);

    gat_gemm_wmma<<<(n_nodes + 127) / 128, 256, 0, stream>>>(feat, W_w, W_b, h, n_nodes);

    gat_scores<<<(n_nodes + 255) / 256, 256, 0, stream>>>(h, attn_w, s_src, s_dst, n_nodes);

    int eb = (n_edges + 255) / 256;
    gat_edge_max<<<eb, 256, 0, stream>>>(s_src, s_dst, src, dst, attn_b, e_buf, m_bits, n_edges);
    gat_edge_exp<<<eb, 256, 0, stream>>>(e_buf, dst, m_bits, denom, n_edges);
    gat_alpha<<<eb, 256, 0, stream>>>(e_buf, denom, dst, n_edges);

    long long total = (long long)n_edges * OUT_F;
    gat_scatter<<<(int)((total + 255) / 256), 256, 0, stream>>>(h, src, dst, e_buf, out, total);
}